// Normalize_19327352832060
// MI455X (gfx1250) — compile-verified
//
#include <hip/hip_runtime.h>
#include <hip/hip_bf16.h>

// out[i,j] = rsqrt(deg[i]) * A[i,j] * rsqrt(deg[j]),  deg = rowsum(A), N=8192 fp32.
// Pure bandwidth problem: ~768 MB min traffic @ 23.3 TB/s ~= 33 us floor.
// Pass 2 runs in REVERSE order so it starts on the A rows still resident in the
// 192MB L2 from pass 1; NT stores keep the output stream from evicting them.

#define NN       8192
#define ROW_VEC  (NN / 4)     // 2048 float4 per row
#define THREADS  256          // 8 waves of 32 on CDNA5

typedef float v4f __attribute__((ext_vector_type(4)));

// ---------------- Pass 1: deg[i] = sum_j A[i,j];  dinv[i] = rsqrt(deg[i]) ----
__global__ __launch_bounds__(THREADS) void k_rowsum_rsqrt(
    const float* __restrict__ A, float* __restrict__ dinv)
{
    const int row = blockIdx.x;                       // one block per row
    const v4f* __restrict__ Arow =
        reinterpret_cast<const v4f*>(A) + (size_t)row * ROW_VEC;
    const int t = threadIdx.x;

    // 256 threads * 8 float4 = 8192 floats (one row); coalesced b128 loads.
    v4f acc = (v4f)(0.0f);
#pragma unroll
    for (int k = 0; k < 8; ++k) {
        if (k < 6)  // gfx1250 global_prefetch_b8 on the upcoming chunk
            __builtin_prefetch(&Arow[t + (k + 2) * THREADS], 0, 0);
        v4f v = Arow[t + k * THREADS];
        acc += v;
    }
    float s = (acc.x + acc.y) + (acc.z + acc.w);

    // wave32 reduction (warpSize == 32 on gfx1250 — do NOT assume 64)
#pragma unroll
    for (int off = 16; off > 0; off >>= 1)
        s += __shfl_down(s, off, 32);

    __shared__ float red[THREADS / 32];
    const int wave = t >> 5;
    if ((t & 31) == 0) red[wave] = s;
    __syncthreads();
    if (t == 0) {
        float tot = 0.0f;
#pragma unroll
        for (int w = 0; w < THREADS / 32; ++w) tot += red[w];
        dinv[row] = rsqrtf(tot);                      // v_rsq_f32
    }
}

// ---------------- Pass 2: out[i,j] = dinv[i] * A[i,j] * dinv[j] --------------
__global__ __launch_bounds__(THREADS) void k_scale(
    const v4f* __restrict__ A4, const float* __restrict__ dinv,
    v4f* __restrict__ O4)
{
    // Reverse block order: start on rows that are hot in L2 from pass 1.
    const unsigned bid = gridDim.x - 1u - blockIdx.x;
    const unsigned idx = bid * THREADS + threadIdx.x; // float4 index, < 2^24
    const unsigned i   = idx >> 11;                   // row (2048 float4 / row)
    const unsigned jv  = idx & 2047u;                 // float4 column

    // A line is dead after this read -> NT load (still hits L2 if resident).
    v4f a = __builtin_nontemporal_load(&A4[idx]);

    const v4f* D4 = reinterpret_cast<const v4f*>(dinv);
    v4f   dj = D4[jv];        // 32KB table, stays in cache
    float di = dinv[i];       // uniform per block -> scalarized

    v4f r;
    r.x = di * a.x * dj.x;
    r.y = di * a.y * dj.y;
    r.z = di * a.z * dj.z;
    r.w = di * a.w * dj.w;

    // NT store: don't let the 256MB output stream evict not-yet-read A from L2.
    __builtin_nontemporal_store(r, &O4[idx]);
}

extern "C" void kernel_launch(void* const* d_in, const int* in_sizes, int n_in,
                              void* d_out, int out_size, void* d_ws, size_t ws_size,
                              hipStream_t stream) {
    const float* A    = (const float*)d_in[0];
    float*       out  = (float*)d_out;
    float*       dinv = (float*)d_ws;                 // 8192 floats = 32KB scratch

    k_rowsum_rsqrt<<<NN, THREADS, 0, stream>>>(A, dinv);

    const unsigned total4 = (unsigned)((size_t)NN * NN / 4); // 16,777,216
    k_scale<<<total4 / THREADS, THREADS, 0, stream>>>(
        reinterpret_cast<const v4f*>(A), dinv, reinterpret_cast<v4f*>(out));
}